// HyperbolicPostHocQuantizer_21449066676758
// MI455X (gfx1250) — compile-verified
//
#include <hip/hip_runtime.h>

// Hyperbolic VQ EMA update for MI455X (gfx1250, wave32, WMMA + TDM).
// Outputs (flat, in return order): indices[N] (int32 bits), new_weights[K*D],
// new_cluster_size[K], new_ema_w[K*D].

#define N_PTS   65536
#define K_CODES 1024
#define DIMS    64
#define DECAY   0.99f

typedef __attribute__((ext_vector_type(2))) float        v2f;
typedef __attribute__((ext_vector_type(8))) float        v8f;
typedef __attribute__((ext_vector_type(4))) unsigned int u32x4;
typedef __attribute__((ext_vector_type(4))) int          i32x4;
typedef __attribute__((ext_vector_type(8))) int          i32x8;

// ---------------------------------------------------------------------------
// Kernel 1: Poincare nearest-code assignment.
//  - Codebook (1024x64 f32 = 256KB) staged into LDS by the Tensor Data Mover
//    (single tensor_load_to_lds from wave 0, TENSORcnt wait, then barrier).
//  - One wave owns 16 z rows; 64 k-tiles of 16 codes; per tile a chain of 16
//    V_WMMA_F32_16X16X4_F32 with double-buffered B fragments from LDS.
// ---------------------------------------------------------------------------
__global__ void __launch_bounds__(256) hq_assign_kernel(
    const float* __restrict__ z, const float* __restrict__ w,
    int* __restrict__ idx_out) {
  extern __shared__ float lds[];                 // [K*D] codebook + [K] norms
  float* lds_w  = lds;
  float* lds_w2 = lds + K_CODES * DIMS;

  const int tid = threadIdx.x;

  // ---- Stage codebook via TDM (wave 0 issues, everyone waits at barrier) ---
  if (tid < 32) {
    const unsigned long long ga = (unsigned long long)(uintptr_t)w;
    u32x4 g0;
    g0.x = 1u;                                   // count=1, user descriptor
    g0.y = 0u;                                   // lds_addr = 0 (dyn-LDS base)
    g0.z = (unsigned int)ga;                     // global_addr[31:0]
    g0.w = (unsigned int)((ga >> 32) & 0x1FFFFFFu) | (2u << 30); // addr|type=2
    i32x8 g1;
    g1[0] = (int)(2u << 16);                     // data_size = 4 bytes
    g1[1] = (int)((unsigned)DIMS << 16);         // tensor_dim0 = 64 (lo16)
    g1[2] = (int)(((unsigned)K_CODES & 0xFFFFu) << 16); // tensor_dim1 = 1024
    g1[3] = (int)((unsigned)DIMS << 16);         // tile_dim0 = 64
    g1[4] = K_CODES;                             // tile_dim1 = 1024, tile_dim2=0
    g1[5] = DIMS;                                // tensor_dim0_stride = 64 (lo32)
    g1[6] = 0;
    g1[7] = 0;
    i32x4 g2 = {0, 0, 0, 0};
    i32x4 g3 = {0, 0, 0, 0};
    i32x8 g4 = {0, 0, 0, 0, 0, 0, 0, 0};
    __builtin_amdgcn_tensor_load_to_lds(g0, g1, g2, g3, g4, 0);
    __builtin_amdgcn_s_wait_tensorcnt(0);
  }
  __syncthreads();

  // Per-code squared norms from LDS.
  for (int k = tid; k < K_CODES; k += 256) {
    const float* row = lds_w + k * DIMS;
    float s = 0.f;
    #pragma unroll
    for (int d = 0; d < DIMS; ++d) s += row[d] * row[d];
    lds_w2[k] = s;
  }
  __syncthreads();

  const int wave = tid >> 5;
  const int lane = tid & 31;
  const int n16  = lane & 15;        // column / candidate slot within tile
  const int hi   = lane >> 4;        // which K-inner pair this lane holds
  const int off  = hi * 2;           // d-offset within each 4-chunk
  const int m0   = (blockIdx.x * 8 + wave) * 16;   // first z row of this wave

  // Preload A fragments: this lane's half of row (m0+n16), 16 x v2f = 32 f32.
  v2f a[16];
  float sq = 0.f;
  {
    const float* zrow = z + (size_t)(m0 + n16) * DIMS;
    #pragma unroll
    for (int t = 0; t < 16; ++t) {
      a[t] = *reinterpret_cast<const v2f*>(zrow + t * 4 + off);
      sq += a[t].x * a[t].x + a[t].y * a[t].y;
    }
  }
  sq += __shfl_xor(sq, 16);          // full ||z_row||^2, row = m0 + n16
  float z2m[8];
  #pragma unroll
  for (int j = 0; j < 8; ++j) z2m[j] = __shfl(sq, hi * 8 + j);

  float best[8];
  int   bestk[8];
  #pragma unroll
  for (int j = 0; j < 8; ++j) { best[j] = 3.4e38f; bestk[j] = 0; }

  // B-fragment loader (16 codes x 4 K-inner for this lane's half).
  auto loadB = [&](v2f (&b)[16], int k0) {
    const float* wrow = lds_w + (size_t)(k0 + n16) * DIMS;
    #pragma unroll
    for (int t = 0; t < 16; ++t)
      b[t] = *reinterpret_cast<const v2f*>(wrow + t * 4 + off);
  };
  // One 16x16 tile: WMMA chain + Poincare score + running argmin.
  auto tile = [&](const v2f (&b)[16], int k0) {
    v8f c = {};
    #pragma unroll
    for (int t = 0; t < 16; ++t)
      c = __builtin_amdgcn_wmma_f32_16x16x4_f32(
          false, a[t], false, b[t], (short)0, c, false, false);
    const float w2v    = lds_w2[k0 + n16];
    const float inv_rv = __builtin_amdgcn_rcpf(1.0f - w2v);
    const int   kc     = k0 + n16;
    #pragma unroll
    for (int j = 0; j < 8; ++j) {
      float score = (z2m[j] + w2v - 2.0f * c[j]) * inv_rv;
      if (score < best[j]) { best[j] = score; bestk[j] = kc; }
    }
  };

  // Software-pipelined sweep over all K codes (double-buffered B).
  v2f b0[16], b1[16];
  loadB(b0, 0);
  for (int k0 = 0; k0 < K_CODES; k0 += 32) {
    loadB(b1, k0 + 16);
    tile(b0, k0);
    if (k0 + 32 < K_CODES) loadB(b0, k0 + 32);
    tile(b1, k0 + 16);
  }

  // Argmin across the 16 candidate lanes of this half (bit 4 untouched).
  #pragma unroll
  for (int mask = 8; mask >= 1; mask >>= 1) {
    #pragma unroll
    for (int j = 0; j < 8; ++j) {
      float ob = __shfl_xor(best[j],  mask);
      int   ok = __shfl_xor(bestk[j], mask);
      if (ob < best[j] || (ob == best[j] && ok < bestk[j])) {
        best[j] = ob; bestk[j] = ok;
      }
    }
  }
  if (n16 == 0) {
    #pragma unroll
    for (int j = 0; j < 8; ++j) idx_out[m0 + hi * 8 + j] = bestk[j];
  }
}

// ---------------------------------------------------------------------------
// Kernel 2: segment sums via global float atomics. N*16 threads, 4 dims each.
// ---------------------------------------------------------------------------
__global__ void __launch_bounds__(256) hq_scatter_kernel(
    const float* __restrict__ z, const int* __restrict__ idx,
    float* __restrict__ sum_kd, float* __restrict__ cnt_k) {
  const int t = blockIdx.x * 256 + threadIdx.x;
  const int n = t >> 4;
  const int g = t & 15;
  const int k = idx[n];
  const float* zp = z + (size_t)n * DIMS + g * 4;
  float* sp = sum_kd + (size_t)k * DIMS + g * 4;
  #pragma unroll
  for (int i = 0; i < 4; ++i) atomicAdd(sp + i, zp[i]);
  if (g == 0) atomicAdd(cnt_k + k, 1.0f);
}

// ---------------------------------------------------------------------------
// Kernel 3: EMA update + Poincare-ball renorm. One 64-thread block per code.
// ---------------------------------------------------------------------------
__global__ void __launch_bounds__(64) hq_ema_kernel(
    const float* __restrict__ ema_w, const float* __restrict__ cs,
    const float* __restrict__ sum_kd, const float* __restrict__ cnt_k,
    float* __restrict__ w_out, float* __restrict__ cs_out,
    float* __restrict__ ema_out) {
  __shared__ float red[64];
  const int k = blockIdx.x;
  const int d = threadIdx.x;
  const float e  = ema_w[(size_t)k * DIMS + d] * DECAY +
                   (1.0f - DECAY) * sum_kd[(size_t)k * DIMS + d];
  const float c  = cs[k] * DECAY + (1.0f - DECAY) * cnt_k[k];
  const float wv = e / (c + 1e-5f);
  red[d] = wv * wv;
  __syncthreads();
  for (int s = 32; s >= 1; s >>= 1) {
    if (d < s) red[d] += red[d + s];
    __syncthreads();
  }
  const float norm  = sqrtf(red[0]);
  const float scale = fminf(norm, 0.999f) / (norm + 1e-10f);
  w_out[(size_t)k * DIMS + d]   = wv * scale;
  ema_out[(size_t)k * DIMS + d] = e;
  if (d == 0) cs_out[k] = c;
}

// ---------------------------------------------------------------------------
extern "C" void kernel_launch(void* const* d_in, const int* in_sizes, int n_in,
                              void* d_out, int out_size, void* d_ws,
                              size_t ws_size, hipStream_t stream) {
  const float* z      = (const float*)d_in[0];   // [N, D]
  const float* weight = (const float*)d_in[1];   // [K, D]
  const float* csize  = (const float*)d_in[2];   // [K]
  const float* ema_w  = (const float*)d_in[3];   // [K, D]

  float* out     = (float*)d_out;
  int*   idx_out = (int*)out;                            // [N] (int32 bits)
  float* w_out   = out + N_PTS;                          // [K*D]
  float* cs_out  = w_out + K_CODES * DIMS;               // [K]
  float* ema_out = cs_out + K_CODES;                     // [K*D]

  float* sum_kd = (float*)d_ws;                          // [K*D]
  float* cnt_k  = sum_kd + K_CODES * DIMS;               // [K]

  (void)hipMemsetAsync(d_ws, 0,
                       (size_t)(K_CODES * DIMS + K_CODES) * sizeof(float),
                       stream);

  const size_t ldsBytes = (size_t)(K_CODES * DIMS + K_CODES) * sizeof(float);
  (void)hipFuncSetAttribute(reinterpret_cast<const void*>(hq_assign_kernel),
                            hipFuncAttributeMaxDynamicSharedMemorySize,
                            (int)ldsBytes);

  hq_assign_kernel<<<N_PTS / 128, 256, ldsBytes, stream>>>(z, weight, idx_out);
  hq_scatter_kernel<<<(N_PTS * 16) / 256, 256, 0, stream>>>(z, idx_out, sum_kd,
                                                            cnt_k);
  hq_ema_kernel<<<K_CODES, DIMS, 0, stream>>>(ema_w, csize, sum_kd, cnt_k,
                                              w_out, cs_out, ema_out);
}